// GCNDiehlq2_22763326669149
// MI455X (gfx1250) — compile-verified
//
#include <hip/hip_runtime.h>
#include <hip/hip_bf16.h>
#include <math.h>

// Problem constants (match reference)
#define NN 50000
#define NE 800000
#define FD 128
#define NG 128

typedef float v8f __attribute__((ext_vector_type(8)));
typedef float v2f __attribute__((ext_vector_type(2)));

// Only probe for the device builtin during the device pass; the host pass
// doesn't declare amdgcn builtins and would trip the #error spuriously.
#if defined(__HIP_DEVICE_COMPILE__)
#if !__has_builtin(__builtin_amdgcn_wmma_f32_16x16x4_f32)
#error "missing __builtin_amdgcn_wmma_f32_16x16x4_f32 on this toolchain (device pass)"
#endif
#endif

// ---------------------------------------------------------------------------
// Utility kernels
// ---------------------------------------------------------------------------

__global__ void k_zero_f4(float4* __restrict__ p, int n4) {
    int i = blockIdx.x * blockDim.x + threadIdx.x;
    if (i < n4) p[i] = make_float4(0.f, 0.f, 0.f, 0.f);
}

// deg[dst]++ over edges ; cnt[batch[n]]++ over nodes (fused)
__global__ void k_deg_cnt(const int* __restrict__ edges, const int* __restrict__ batch,
                          float* __restrict__ deg, float* __restrict__ cnt) {
    int i = blockIdx.x * blockDim.x + threadIdx.x;
    if (i < NE) {
        int dst = edges[2 * i + 1];
        atomicAdd(deg + dst, 1.0f);
    }
    if (i < NN) {
        int b = batch[i];
        atomicAdd(cnt + b, 1.0f);
    }
}

// p[i] = 1 / max(p[i], 1)
__global__ void k_invert(float* __restrict__ p, int n) {
    int i = blockIdx.x * blockDim.x + threadIdx.x;
    if (i < n) p[i] = 1.0f / fmaxf(p[i], 1.0f);
}

// agg[dst,:] += h[src,:]  — one thread per (edge, 4-feature quad).
// agg (25.6MB) lives in L2 (192MB) so these f32 atomics resolve in L2.
__global__ void k_scatter(const float* __restrict__ h, const int* __restrict__ edges,
                          float* __restrict__ agg) {
    long long t = (long long)blockIdx.x * blockDim.x + threadIdx.x;
    if (t >= (long long)NE * 32) return;
    int e = (int)(t >> 5);
    int q = ((int)t & 31) << 2;
    int src = edges[2 * e + 0];
    int dst = edges[2 * e + 1];
    float4 v = *(const float4*)(h + (long long)src * FD + q);
    float* a = agg + (long long)dst * FD + q;
    atomicAdd(a + 0, v.x);
    atomicAdd(a + 1, v.y);
    atomicAdd(a + 2, v.z);
    atomicAdd(a + 3, v.w);
}

// g[batch[n],:] += h[n,:]  — one thread per (node, 4-feature quad).
__global__ void k_pool(const float* __restrict__ h, const int* __restrict__ batch,
                       float* __restrict__ g) {
    long long t = (long long)blockIdx.x * blockDim.x + threadIdx.x;
    if (t >= (long long)NN * 32) return;
    int n = (int)(t >> 5);
    int q = ((int)t & 31) << 2;
    int b = batch[n];
    float4 v = *(const float4*)(h + (long long)n * FD + q);
    float* gp = g + (long long)b * FD + q;
    atomicAdd(gp + 0, v.x);
    atomicAdd(gp + 1, v.y);
    atomicAdd(gp + 2, v.z);
    atomicAdd(gp + 3, v.w);
}

// ---------------------------------------------------------------------------
// SAGE layer GEMM:  out = relu( (agg * invdeg[m]) @ Wl + h @ Wr + b )
// Block = 16 rows x 128 cols. 8 waves; wave w computes the 16x16 tile at
// n0 = 16*w using V_WMMA_F32_16X16X4_F32, accumulating over K=128 (32 steps
// per operand matrix, 64 WMMAs per tile).
// A-panels staged in LDS with pitch 132 floats (128+4) -> conflict-free
// strided A-fragment reads (4*m mod 64 distinct for m=0..15).
// ---------------------------------------------------------------------------
#define APITCH 132

__global__ __launch_bounds__(256) void k_sage_gemm(
        const float* __restrict__ agg, const float* __restrict__ invdeg,
        const float* __restrict__ h,
        const float* __restrict__ Wl, const float* __restrict__ Wr,
        const float* __restrict__ bias, float* __restrict__ out) {
    __shared__ float sA[16 * APITCH];
    __shared__ float sH[16 * APITCH];

    const int m0 = blockIdx.x * 16;

    // Stage 16x128 panels of deg-scaled agg and of h (coalesced float4 loads).
    for (int i = threadIdx.x; i < 16 * 32; i += 256) {
        int row = i >> 5;
        int c   = (i & 31) << 2;
        float s = invdeg[m0 + row];
        float4 a = *(const float4*)(agg + (long long)(m0 + row) * FD + c);
        float4 v = make_float4(a.x * s, a.y * s, a.z * s, a.w * s);
        *(float4*)(sA + row * APITCH + c) = v;
        *(float4*)(sH + row * APITCH + c) = *(const float4*)(h + (long long)(m0 + row) * FD + c);
    }
    __syncthreads();

    const int lane = threadIdx.x & 31;
    const int wave = threadIdx.x >> 5;
    const int n0   = wave * 16;
    const int m    = lane & 15;          // row within A fragment / col within B,C,D
    const int koff = (lane >> 4) * 2;    // half-wave K offset

    v8f acc = {0.f, 0.f, 0.f, 0.f, 0.f, 0.f, 0.f, 0.f};

    for (int k = 0; k < FD; k += 4) {
        int kk = k + koff;
        v2f aA, aH, bL, bR;
        aA.x = sA[m * APITCH + kk];
        aA.y = sA[m * APITCH + kk + 1];
        aH.x = sH[m * APITCH + kk];
        aH.y = sH[m * APITCH + kk + 1];
        // W is (din=128, dout=128) row-major: W[k][n] = W[k*128 + n]
        bL.x = Wl[kk * FD + n0 + m];
        bL.y = Wl[(kk + 1) * FD + n0 + m];
        bR.x = Wr[kk * FD + n0 + m];
        bR.y = Wr[(kk + 1) * FD + n0 + m];
        acc = __builtin_amdgcn_wmma_f32_16x16x4_f32(false, aA, false, bL,
                                                    (short)0, acc, false, false);
        acc = __builtin_amdgcn_wmma_f32_16x16x4_f32(false, aH, false, bR,
                                                    (short)0, acc, false, false);
    }

    // Epilogue: C/D layout — VGPR r: lanes 0-15 -> M=r, lanes 16-31 -> M=r+8
    const int half = lane >> 4;
    const int n    = n0 + (lane & 15);
    const float bn = bias[n];
    #pragma unroll
    for (int r = 0; r < 8; ++r) {
        int mm = r + half * 8;
        float v = acc[r] + bn;
        out[(long long)(m0 + mm) * FD + n] = fmaxf(v, 0.f);
    }
}

// ---------------------------------------------------------------------------
// Pooled MLP GEMM: out = relu( (A * scale[m]) @ W + b ), A is 128x128.
// Same WMMA structure, single input matrix. Grid = 8 blocks (128/16 m-tiles).
// ---------------------------------------------------------------------------
__global__ __launch_bounds__(256) void k_scaled_gemm_relu(
        const float* __restrict__ A, const float* __restrict__ scale,
        const float* __restrict__ W, const float* __restrict__ bias,
        float* __restrict__ out) {
    __shared__ float sA[16 * APITCH];
    const int m0 = blockIdx.x * 16;

    for (int i = threadIdx.x; i < 16 * 32; i += 256) {
        int row = i >> 5;
        int c   = (i & 31) << 2;
        float s = scale[m0 + row];
        float4 a = *(const float4*)(A + (long long)(m0 + row) * FD + c);
        *(float4*)(sA + row * APITCH + c) = make_float4(a.x * s, a.y * s, a.z * s, a.w * s);
    }
    __syncthreads();

    const int lane = threadIdx.x & 31;
    const int wave = threadIdx.x >> 5;
    const int n0   = wave * 16;
    const int m    = lane & 15;
    const int koff = (lane >> 4) * 2;

    v8f acc = {0.f, 0.f, 0.f, 0.f, 0.f, 0.f, 0.f, 0.f};
    for (int k = 0; k < FD; k += 4) {
        int kk = k + koff;
        v2f a, b;
        a.x = sA[m * APITCH + kk];
        a.y = sA[m * APITCH + kk + 1];
        b.x = W[kk * FD + n0 + m];
        b.y = W[(kk + 1) * FD + n0 + m];
        acc = __builtin_amdgcn_wmma_f32_16x16x4_f32(false, a, false, b,
                                                    (short)0, acc, false, false);
    }

    const int half = lane >> 4;
    const int n    = n0 + (lane & 15);
    const float bn = bias[n];
    #pragma unroll
    for (int r = 0; r < 8; ++r) {
        int mm = r + half * 8;
        out[(long long)(m0 + mm) * FD + n] = fmaxf(acc[r] + bn, 0.f);
    }
}

// Final head: out[i] = sigmoid( dot(g2[i,:], Wf2) + bf2 ), i in [0,128)
__global__ void k_head(const float* __restrict__ g2, const float* __restrict__ Wf2,
                       const float* __restrict__ bf2, float* __restrict__ out) {
    int i = blockIdx.x * blockDim.x + threadIdx.x;
    if (i >= NG) return;
    float acc = bf2[0];
    const float* row = g2 + (long long)i * FD;
    #pragma unroll 8
    for (int k = 0; k < FD; ++k) acc += row[k] * Wf2[k];
    out[i] = 1.0f / (1.0f + __expf(-acc));
}

// ---------------------------------------------------------------------------
// Launch
// ---------------------------------------------------------------------------
extern "C" void kernel_launch(void* const* d_in, const int* in_sizes, int n_in,
                              void* d_out, int out_size, void* d_ws, size_t ws_size,
                              hipStream_t stream) {
    const float* x     = (const float*)d_in[0];
    const int* edges   = (const int*)d_in[1];
    const int* batch   = (const int*)d_in[2];
    const float* Wl[5] = {(const float*)d_in[3], (const float*)d_in[6], (const float*)d_in[9],
                          (const float*)d_in[12], (const float*)d_in[15]};
    const float* Wr[5] = {(const float*)d_in[4], (const float*)d_in[7], (const float*)d_in[10],
                          (const float*)d_in[13], (const float*)d_in[16]};
    const float* bb[5] = {(const float*)d_in[5], (const float*)d_in[8], (const float*)d_in[11],
                          (const float*)d_in[14], (const float*)d_in[17]};
    const float* Wf1 = (const float*)d_in[18];
    const float* bf1 = (const float*)d_in[19];
    const float* Wf2 = (const float*)d_in[20];
    const float* bf2 = (const float*)d_in[21];
    float* out = (float*)d_out;

    // Workspace layout (floats)
    float* ws = (float*)d_ws;
    float* deg  = ws;                         // NN
    float* cnt  = deg + NN;                   // NG
    float* agg  = cnt + NG;                   // NN*FD
    float* h0   = agg + (size_t)NN * FD;      // NN*FD
    float* h1   = h0  + (size_t)NN * FD;      // NN*FD
    float* g    = h1  + (size_t)NN * FD;      // NG*FD
    float* g2   = g   + (size_t)NG * FD;      // NG*FD

    const int T = 256;
    auto blocks = [](long long n, int t) { return (int)((n + t - 1) / t); };

    // deg / cnt (zero then accumulate then invert)
    k_zero_f4<<<blocks((NN + NG) / 4, T), T, 0, stream>>>((float4*)deg, (NN + NG) / 4);
    k_deg_cnt<<<blocks(NE, T), T, 0, stream>>>(edges, batch, deg, cnt);
    k_invert<<<blocks(NN + NG, T), T, 0, stream>>>(deg, NN + NG); // deg & cnt contiguous

    const float* hin = x;
    float* bufs[2] = {h0, h1};
    for (int l = 0; l < 5; ++l) {
        float* hout = bufs[l & 1];
        k_zero_f4<<<blocks((long long)NN * FD / 4, T), T, 0, stream>>>(
            (float4*)agg, NN * FD / 4);
        k_scatter<<<blocks((long long)NE * 32, T), T, 0, stream>>>(hin, edges, agg);
        k_sage_gemm<<<NN / 16, T, 0, stream>>>(agg, deg, hin, Wl[l], Wr[l], bb[l], hout);
        hin = hout;
    }

    // Graph pooling + MLP head
    k_zero_f4<<<blocks((long long)NG * FD / 4, T), T, 0, stream>>>((float4*)g, NG * FD / 4);
    k_pool<<<blocks((long long)NN * 32, T), T, 0, stream>>>(hin, batch, g);
    k_scaled_gemm_relu<<<NG / 16, T, 0, stream>>>(g, cnt, Wf1, bf1, g2);
    k_head<<<1, NG, 0, stream>>>(g2, Wf2, bf2, out);
}